// TDrumorGCN_34634616275004
// MI455X (gfx1250) — compile-verified
//
#include <hip/hip_runtime.h>

typedef __attribute__((ext_vector_type(2))) float v2f;
typedef __attribute__((ext_vector_type(8))) float v8f;

#define HIDC 64  // both conv layers output 64 channels

// ---------------------------------------------------------------- degrees ---
__global__ void deg_init_kernel(float* __restrict__ deg, int n) {
  int i = blockIdx.x * blockDim.x + threadIdx.x;
  if (i < n) deg[i] = 1.0f;  // self loop contributes 1 to every node's degree
}

__global__ void deg_scatter_kernel(const int* __restrict__ dst,
                                   float* __restrict__ deg, int E) {
  int i = blockIdx.x * blockDim.x + threadIdx.x;
  int stride = gridDim.x * blockDim.x;
  for (int e = i; e < E; e += stride) atomicAdd(&deg[dst[e]], 1.0f);
}

__global__ void dinv_kernel(float* __restrict__ deg, int n) {
  int i = blockIdx.x * blockDim.x + threadIdx.x;
  if (i < n) {
    float d = deg[i];
    deg[i] = d > 0.0f ? rsqrtf(d) : 0.0f;  // in-place: deg -> dinv
  }
}

// ------------------------------------------------------- WMMA fp32 GEMM -----
// C[n,64] = A[n,K] @ W[K,64], fp32, using V_WMMA_F32_16X16X4_F32.
// One wave computes a 16x64 output stripe: 4 column tiles of 16x16.
// A-frag layout (16x4 f32): lanes 0-15 hold K={k,k+1}, lanes 16-31 K={k+2,k+3}
// B-frag layout (4x16 f32): mirrored (lane group selects K pair, lane%16 = N)
// D layout (16x16 f32, 8 VGPRs): vgpr r -> M = r + 8*(lane>=16), N = lane%16
template <int K>
__launch_bounds__(256)
__global__ void gemm_wmma_kernel(const float* __restrict__ A,
                                 const float* __restrict__ W,
                                 float* __restrict__ C, int n) {
  const int wave = threadIdx.x >> 5;
  const int lane = threadIdx.x & 31;
  const int rowBase = (blockIdx.x * 8 + wave) << 4;
  if (rowBase >= n) return;  // whole-wave uniform exit; EXEC stays all-1s
  const int lhalf = lane >> 4;
  const int lmod = lane & 15;

  // Preload all A fragments for this wave's 16 rows (K/4 frags, regs only)
  v2f afrag[K / 4];
  const float* arow = A + (size_t)(rowBase + lmod) * K;
#pragma unroll
  for (int k = 0; k < K / 4; ++k) {
    const int kk = k * 4 + lhalf * 2;
    afrag[k].x = arow[kk];
    afrag[k].y = arow[kk + 1];
  }

#pragma unroll
  for (int nt = 0; nt < 4; ++nt) {
    const int col = nt * 16 + lmod;
    v8f acc = {};
#pragma unroll
    for (int k = 0; k < K / 4; ++k) {
      const int kk = k * 4 + lhalf * 2;
      v2f bfrag;
      bfrag.x = W[kk * HIDC + col];
      bfrag.y = W[(kk + 1) * HIDC + col];
      acc = __builtin_amdgcn_wmma_f32_16x16x4_f32(
          false, afrag[k], false, bfrag, (short)0, acc, false, false);
    }
#pragma unroll
    for (int r = 0; r < 8; ++r) {
      C[(size_t)(rowBase + r + lhalf * 8) * HIDC + col] = acc[r];
    }
  }
}

// ------------------------------------------------- scatter aggregation ------
// agg[n,c] = hpre[n,c]*dinv[n]^2 + b[c]   (self-loop term + bias)
__global__ void agg_init_kernel(const float* __restrict__ hpre,
                                const float* __restrict__ dinv,
                                const float* __restrict__ bias,
                                float* __restrict__ agg, int n) {
  int i = blockIdx.x * blockDim.x + threadIdx.x;
  int total = n * HIDC;
  if (i < total) {
    int node = i >> 6;
    int c = i & 63;
    float di = dinv[node];
    agg[i] = hpre[i] * di * di + bias[c];
  }
}

// agg[dst,c] += hpre[src,c] * dinv[src]*dinv[dst]; wave per edge, 2 ch/lane
__global__ void edge_scatter_kernel(const int* __restrict__ src,
                                    const int* __restrict__ dst,
                                    const float* __restrict__ dinv,
                                    const float* __restrict__ hpre,
                                    float* __restrict__ agg, int E) {
  int wid = (blockIdx.x * blockDim.x + threadIdx.x) >> 5;
  int lane = threadIdx.x & 31;
  int nwaves = (gridDim.x * blockDim.x) >> 5;
  for (int e = wid; e < E; e += nwaves) {
    int s = src[e];
    int d = dst[e];
    float norm = dinv[s] * dinv[d];
    const float* hs = hpre + (size_t)s * HIDC;
    float* ad = agg + (size_t)d * HIDC;
    atomicAdd(&ad[lane], hs[lane] * norm);
    atomicAdd(&ad[lane + 32], hs[lane + 32] * norm);
  }
}

// ------------------------------------------------------ concat + relu -------
// hcat[n, 0:64]  = relu(h1[n,:])
// hcat[n,64:192] = relu(root[batch[n],:])      (block per node, 192 threads)
__global__ void hcat_kernel(const float* __restrict__ h1,
                            const float* __restrict__ root,
                            const int* __restrict__ batch,
                            float* __restrict__ hcat) {
  int node = blockIdx.x;
  int c = threadIdx.x;  // 0..191
  float v;
  if (c < 64)
    v = h1[(size_t)node * 64 + c];
  else
    v = root[(size_t)batch[node] * 128 + (c - 64)];
  hcat[(size_t)node * 192 + c] = fmaxf(v, 0.0f);
}

// ------------------------------------------------------------- pooling ------
__global__ void pool_zero_kernel(float* __restrict__ p, int count) {
  int i = blockIdx.x * blockDim.x + threadIdx.x;
  if (i < count) p[i] = 0.0f;
}

// psum[g,c] += relu(agg2[n,c]); pcnt[g] += 1   (wave per node, 2 ch/lane)
__global__ void pool_kernel(const float* __restrict__ agg2,
                            const int* __restrict__ batch,
                            float* __restrict__ psum,
                            float* __restrict__ pcnt, int n) {
  int wid = (blockIdx.x * blockDim.x + threadIdx.x) >> 5;
  int lane = threadIdx.x & 31;
  int nwaves = (gridDim.x * blockDim.x) >> 5;
  for (int node = wid; node < n; node += nwaves) {
    int g = batch[node];
    const float* row = agg2 + (size_t)node * HIDC;
    atomicAdd(&psum[(size_t)g * HIDC + lane], fmaxf(row[lane], 0.0f));
    atomicAdd(&psum[(size_t)g * HIDC + lane + 32], fmaxf(row[lane + 32], 0.0f));
    if (lane == 0) atomicAdd(&pcnt[g], 1.0f);
  }
}

// out[g, 0:64]  = psum[g,:]/pcnt[g]
// out[g,64:128] = h1[rootindex[g], :]
__global__ void finalize_kernel(const float* __restrict__ psum,
                                const float* __restrict__ pcnt,
                                const float* __restrict__ h1,
                                const int* __restrict__ rootindex,
                                float* __restrict__ out, int G) {
  int i = blockIdx.x * blockDim.x + threadIdx.x;
  if (i < G * 128) {
    int g = i >> 7;
    int c = i & 127;
    float v;
    if (c < 64)
      v = psum[(size_t)g * 64 + c] / pcnt[g];
    else
      v = h1[(size_t)rootindex[g] * 64 + (c - 64)];
    out[i] = v;
  }
}

// ---------------------------------------------------------------- launch ----
extern "C" void kernel_launch(void* const* d_in, const int* in_sizes, int n_in,
                              void* d_out, int out_size, void* d_ws,
                              size_t ws_size, hipStream_t stream) {
  const float* x    = (const float*)d_in[0];   // [N,128]
  const int*   eidx = (const int*)d_in[1];     // [2,E]
  const int*   batch = (const int*)d_in[2];    // [N]
  const int*   rooti = (const int*)d_in[3];    // [G]
  const float* root = (const float*)d_in[4];   // [G,128]
  const float* W1   = (const float*)d_in[5];   // [128,64]
  const float* b1   = (const float*)d_in[6];   // [64]
  const float* W2   = (const float*)d_in[7];   // [192,64]
  const float* b2   = (const float*)d_in[8];   // [64]
  float* out = (float*)d_out;

  const int N = in_sizes[0] / 128;
  const int E = in_sizes[1] / 2;
  const int G = in_sizes[3];
  const int* src = eidx;
  const int* dst = eidx + E;

  // workspace layout (floats); agg2 aliases hcat (stream-serialized, safe)
  float* ws   = (float*)d_ws;
  float* dinv = ws;                            // [N]
  float* h1   = dinv + N;                      // [N,64]  conv1 out (pre-relu)
  float* bufA = h1 + (size_t)N * 64;           // [N,64]  h_pre / p2
  float* hcat = bufA + (size_t)N * 64;         // [N,192] concat input
  float* agg2 = hcat;                          // [N,64]  alias
  float* psum = hcat + (size_t)N * 192;        // [G,64]
  float* pcnt = psum + (size_t)G * 64;         // [G]

  // 1. symmetric-norm degrees
  deg_init_kernel<<<(N + 255) / 256, 256, 0, stream>>>(dinv, N);
  deg_scatter_kernel<<<1024, 256, 0, stream>>>(dst, dinv, E);
  dinv_kernel<<<(N + 255) / 256, 256, 0, stream>>>(dinv, N);

  // 2. conv1: GEMM (WMMA fp32) + self-loop/bias init + edge scatter
  gemm_wmma_kernel<128><<<(N + 127) / 128, 256, 0, stream>>>(x, W1, bufA, N);
  agg_init_kernel<<<(N * 64 + 255) / 256, 256, 0, stream>>>(bufA, dinv, b1, h1, N);
  edge_scatter_kernel<<<2048, 256, 0, stream>>>(src, dst, dinv, bufA, h1, E);

  // 3. concat [h1 | root[batch]] + relu
  hcat_kernel<<<N, 192, 0, stream>>>(h1, root, batch, hcat);

  // 4. conv2: GEMM (WMMA fp32, K=192) + init + edge scatter
  gemm_wmma_kernel<192><<<(N + 127) / 128, 256, 0, stream>>>(hcat, W2, bufA, N);
  agg_init_kernel<<<(N * 64 + 255) / 256, 256, 0, stream>>>(bufA, dinv, b2, agg2, N);
  edge_scatter_kernel<<<2048, 256, 0, stream>>>(src, dst, dinv, bufA, agg2, E);

  // 5. relu + scatter-mean pool + root concat
  pool_zero_kernel<<<(G * 65 + 255) / 256, 256, 0, stream>>>(psum, G * 65);
  pool_kernel<<<2048, 256, 0, stream>>>(agg2, batch, psum, pcnt, N);
  finalize_kernel<<<(G * 128 + 255) / 256, 256, 0, stream>>>(psum, pcnt, h1,
                                                             rooti, out, G);
}